// PS_35811437314228
// MI455X (gfx1250) — compile-verified
//
#include <hip/hip_runtime.h>
#include <cstdint>

// Pixel shuffle R=4, N_CHANNEL=3, B=16, H=W=256.
// out[g, i*4+q, j*4+p] = in_group[g][i*4096 + 4*(j*4+p) + q]
// Pure data movement: 403 MB total @ 23.3 TB/s -> ~17.3 us floor.
// Strategy: per (g,i) slab (4096 contiguous input floats = 16 KB):
//   async-DMA the slab into padded LDS (global_load_async_to_lds_b128),
//   s_wait_asynccnt + barrier, then emit 4 output rows as coalesced
//   nontemporal B128 stores, gathering stride-4 from LDS.

typedef float v4f __attribute__((ext_vector_type(4)));
typedef int   v4i __attribute__((__vector_size__(4 * sizeof(int))));

#define AS1 __attribute__((address_space(1)))
#define AS3 __attribute__((address_space(3)))

__device__ __forceinline__ void async_load_b128_to_lds(const v4f* gsrc, float* lds_dst) {
#if __has_builtin(__builtin_amdgcn_global_load_async_to_lds_b128)
    // Signature (from hipcc diagnostic): (AS1 v4i*, AS3 v4i*, imm offset, imm cpol)
    __builtin_amdgcn_global_load_async_to_lds_b128(
        (AS1 v4i*)(gsrc), (AS3 v4i*)(lds_dst), /*imm offset*/0, /*cpol*/0);
#else
    // VDST VGPR = LDS byte offset, VADDR = 64-bit global address, no SADDR.
    uint32_t loff  = (uint32_t)(uintptr_t)(AS3 void*)lds_dst;
    uint64_t gaddr = (uint64_t)(uintptr_t)gsrc;
    asm volatile("global_load_async_to_lds_b128 %0, %1, off"
                 :: "v"(loff), "v"(gaddr)
                 : "memory");
#endif
}

__device__ __forceinline__ void wait_async_zero() {
#if __has_builtin(__builtin_amdgcn_s_wait_asynccnt)
    __builtin_amdgcn_s_wait_asynccnt(0);
#else
    asm volatile("s_wait_asynccnt 0" ::: "memory");
#endif
}

// LDS: 4096 data words + 4 pad words per 64 -> 4352 words (17 KB).
// Padded word index: w(s) = s + 4*(s>>6). Multiples of 4 stay multiples of 4,
// so every B128 LDS access remains 16B-aligned and non-straddling.
#define LDS_WORDS 4352

__global__ __launch_bounds__(256)
void pixel_shuffle_r4(const float* __restrict__ in, float* __restrict__ out) {
    __shared__ float lds[LDS_WORDS];

    const uint32_t tid  = threadIdx.x;          // 0..255
    const uint32_t slab = blockIdx.x;           // g*256 + i  (48*256 slabs)

    const v4f* __restrict__ in4  = reinterpret_cast<const v4f*>(in)  + (size_t)slab * 1024u;
    v4f* __restrict__       out4 = reinterpret_cast<v4f*>(out)       + (size_t)slab * 1024u;

    // ---- Phase 1: stage 16 KB contiguous slab into LDS via async DMA ----
    #pragma unroll
    for (uint32_t k = 0; k < 4u; ++k) {
        const uint32_t idx4 = k * 256u + tid;                 // float4 index in slab [0,1024)
        const uint32_t w    = 4u * idx4 + 4u * (idx4 >> 4);   // padded word of s = 4*idx4
        async_load_b128_to_lds(in4 + idx4, &lds[w]);
    }
    wait_async_zero();        // this wave's async copies have landed in LDS
    __syncthreads();          // all waves' copies visible workgroup-wide

    // ---- Phase 2: emit 4 output rows, one coalesced float4 per thread/row ----
    // Row q, col group c4=tid covers cols 4*tid..4*tid+3:
    //   source s = 16*tid + 4*kk + q  -> padded word = s + 4*(tid>>2)
    #pragma unroll
    for (uint32_t q = 0; q < 4u; ++q) {
        const uint32_t base = 16u * tid + q + 4u * (tid >> 2);
        v4f v;
        v.x = lds[base];
        v.y = lds[base + 4u];
        v.z = lds[base + 8u];
        v.w = lds[base + 12u];
        __builtin_nontemporal_store(v, out4 + (q * 256u + tid));
    }
}

extern "C" void kernel_launch(void* const* d_in, const int* in_sizes, int n_in,
                              void* d_out, int out_size, void* d_ws, size_t ws_size,
                              hipStream_t stream) {
    (void)n_in; (void)out_size; (void)d_ws; (void)ws_size;
    const float* x = (const float*)d_in[0];
    float* out     = (float*)d_out;

    // One block per (group, input-row) slab of 4096 floats: 48*256 = 12288 blocks.
    const int slabs = in_sizes[0] >> 12;
    pixel_shuffle_r4<<<dim3(slabs), dim3(256), 0, stream>>>(x, out);
}